// EMAVectorQuantizer_80736795230962
// MI455X (gfx1250) — compile-verified
//
#include <hip/hip_runtime.h>
#include <hip/hip_bf16.h>
#include <math.h>

typedef __attribute__((ext_vector_type(2))) float v2f;
typedef __attribute__((ext_vector_type(8))) float v8f;

#define NB    16
#define CD    128        // channels == codebook dim D
#define HW    1024       // 32*32
#define NROWS 16384      // NB * HW
#define KCB   1024       // codebook size

// d_out layout (float elements), reference tuple concatenated flat:
#define OFF_Q    0                       // quantized NCHW: 2,097,152
#define OFF_LOSS 2097152                 // scalar
#define OFF_PERP 2097153                 // scalar
#define OFF_IDX  2097154                 // 16384 (indices as float)
#define OFF_ENC  2113538                 // one-hot 16384*1024 = 16,777,216

// ---------------------------------------------------------------- init
__global__ __launch_bounds__(256) void k_init(const float* __restrict__ emb,
                                              float* __restrict__ enorm,
                                              unsigned* __restrict__ counts) {
    int k = blockIdx.x * 256 + threadIdx.x;
    if (k < KCB) {
        const float* row = emb + k * CD;
        float s = 0.f;
        #pragma unroll
        for (int i = 0; i < CD; ++i) s += row[i] * row[i];
        enorm[k]  = s;
        counts[k] = 0u;
    }
}

// ------------------------------------------------- argmin via fp32 WMMA
// One wave handles 16 rows of x against all 1024 codebook entries.
// dist = ||e||^2 - 2 * <x, e>   (||x||^2 is row-constant: irrelevant to argmin)
__global__ __launch_bounds__(256, 1) void k_argmin(const float* __restrict__ x,
                                                   const float* __restrict__ emb,
                                                   const float* __restrict__ enorm,
                                                   int* __restrict__ out_idx,
                                                   unsigned* __restrict__ counts) {
    const int lane = threadIdx.x & 31;
    const int wave = threadIdx.x >> 5;
    const int half = lane >> 4;          // 0: lanes 0-15, 1: lanes 16-31
    const int m    = lane & 15;          // row within 16-row tile (A), col (B/C)
    const int row_base = (blockIdx.x * 8 + wave) * 16;

    // global row for this lane's A fragment; whole tile shares one batch image
    const int n  = row_base + m;
    const int b  = n >> 10;
    const int hw = n & 1023;
    const float* xcol = x + (size_t)b * (CD * HW) + hw;   // element d at xcol[d*HW]

    // A fragments 16x4 f32: VGPR0 = K=(0|2), VGPR1 = K=(1|3) per half-wave.
    // Hoisted: reused across all 64 column tiles.
    v2f a[32];
    #pragma unroll
    for (int kk = 0; kk < 32; ++kk) {
        const int k0 = kk * 4 + half * 2;
        a[kk][0] = xcol[(size_t)k0 * HW];          // lanes 0-15: 16 consecutive hw -> 64B segments
        a[kk][1] = xcol[(size_t)(k0 + 1) * HW];
    }

    float minv[8];
    int   mini[8];
    #pragma unroll
    for (int r = 0; r < 8; ++r) { minv[r] = 3.4e38f; mini[r] = 0; }

    for (int t = 0; t < 64; ++t) {                 // 64 tiles of 16 codebook cols
        // B fragment (4x16 = E^T tile): lane's column is codebook row (t*16+m)
        const float* eb = emb + (size_t)(t * 16 + m) * CD + half * 2;
        v8f c = {0.f, 0.f, 0.f, 0.f, 0.f, 0.f, 0.f, 0.f};
        #pragma unroll
        for (int kk = 0; kk < 32; ++kk) {
            v2f bf = *(const v2f*)(eb + kk * 4);   // 8B-aligned pair (K, K+1)
            c = __builtin_amdgcn_wmma_f32_16x16x4_f32(false, a[kk], false, bf,
                                                      (short)0, c, false, false);
        }
        const int   col = t * 16 + m;
        const float en  = enorm[col];
        #pragma unroll
        for (int r = 0; r < 8; ++r) {              // VGPR r <-> M = r + 8*half
            float dist = en - 2.0f * c[r];
            if (dist < minv[r]) { minv[r] = dist; mini[r] = col; }  // strict <: first col wins ties
        }
    }

    // min-with-index across the 16 lanes (N dimension) of each half-wave
    #pragma unroll
    for (int mask = 8; mask >= 1; mask >>= 1) {
        #pragma unroll
        for (int r = 0; r < 8; ++r) {
            float ov = __shfl_xor(minv[r], mask, 32);
            int   oi = __shfl_xor(mini[r], mask, 32);
            if (ov < minv[r] || (ov == minv[r] && oi < mini[r])) { minv[r] = ov; mini[r] = oi; }
        }
    }

    if (m == 0) {
        #pragma unroll
        for (int r = 0; r < 8; ++r) {
            const int row = row_base + r + half * 8;
            out_idx[row] = mini[r];
            atomicAdd(&counts[mini[r]], 1u);       // integer atomic: deterministic
        }
    }
}

// ------------------------------------------------- zero one-hot region (8B aligned)
__global__ __launch_bounds__(256) void k_zero(float2* __restrict__ p, int n2) {
    int i = blockIdx.x * 256 + threadIdx.x;
    const int stride = gridDim.x * 256;
    const float2 z = make_float2(0.f, 0.f);
    for (; i < n2; i += stride) p[i] = z;
}

// ------------------------------------------------- indices + one-hot ones
__global__ __launch_bounds__(256) void k_scatter(const int* __restrict__ idx,
                                                 float* __restrict__ out) {
    int n = blockIdx.x * 256 + threadIdx.x;
    if (n < NROWS) {
        int k = idx[n];
        out[OFF_IDX + n] = (float)k;
        out[(size_t)OFF_ENC + (size_t)n * KCB + k] = 1.0f;
    }
}

// ------------------------------------------------- quantized out + loss partials
__global__ __launch_bounds__(256) void k_quant(const float* __restrict__ x,
                                               const float* __restrict__ emb,
                                               const int* __restrict__ idx,
                                               float* __restrict__ out,
                                               float* __restrict__ partial) {
    __shared__ float sred[256];
    const int i  = blockIdx.x * 256 + threadIdx.x;   // NCHW flat: ((b*128+d)*1024+hw)
    const int hw = i & 1023;
    const int d  = (i >> 10) & 127;
    const int b  = i >> 17;
    const int n  = (b << 10) | hw;
    const int k  = idx[n];                           // coalesced (consecutive hw)
    const float xv   = x[i];                         // coalesced
    const float q    = emb[k * CD + d];              // scattered gather, L2-resident
    const float diff = q - xv;
    out[i] = xv + diff;                              // straight-through, exact ref arithmetic
    sred[threadIdx.x] = diff * diff;
    __syncthreads();
    for (int s = 128; s > 0; s >>= 1) {
        if (threadIdx.x < s) sred[threadIdx.x] += sred[threadIdx.x + s];
        __syncthreads();
    }
    if (threadIdx.x == 0) partial[blockIdx.x] = sred[0];
}

// ------------------------------------------------- scalars (fixed-order reduce)
__global__ __launch_bounds__(256) void k_final(const float* __restrict__ partial,
                                               const unsigned* __restrict__ counts,
                                               float* __restrict__ out) {
    __shared__ float s1[256];
    float acc = 0.f;
    for (int i = threadIdx.x; i < 8192; i += 256) acc += partial[i];
    s1[threadIdx.x] = acc;
    __syncthreads();
    for (int s = 128; s > 0; s >>= 1) {
        if (threadIdx.x < s) s1[threadIdx.x] += s1[threadIdx.x + s];
        __syncthreads();
    }
    const float loss_total = s1[0];
    __syncthreads();

    float h = 0.f;
    for (int i = threadIdx.x; i < KCB; i += 256) {
        float p = (float)counts[i] / (float)NROWS;
        h -= p * logf(p + 1e-10f);
    }
    s1[threadIdx.x] = h;
    __syncthreads();
    for (int s = 128; s > 0; s >>= 1) {
        if (threadIdx.x < s) s1[threadIdx.x] += s1[threadIdx.x + s];
        __syncthreads();
    }
    if (threadIdx.x == 0) {
        out[OFF_LOSS] = 0.25f * loss_total / (float)(NROWS * CD);
        out[OFF_PERP] = expf(s1[0]);
    }
}

extern "C" void kernel_launch(void* const* d_in, const int* in_sizes, int n_in,
                              void* d_out, int out_size, void* d_ws, size_t ws_size,
                              hipStream_t stream) {
    const float* x   = (const float*)d_in[0];   // (16,128,32,32) f32
    const float* emb = (const float*)d_in[1];   // (1024,128) f32
    float* out = (float*)d_out;

    char*     ws      = (char*)d_ws;
    int*      ws_idx  = (int*)ws;               // 16384 * 4  = 65536 B
    unsigned* counts  = (unsigned*)(ws + 65536);//  1024 * 4  ->  69632
    float*    enorm   = (float*)(ws + 69632);   //  1024 * 4  ->  73728
    float*    partial = (float*)(ws + 73728);   //  8192 * 4  -> 106496

    k_init   <<<   4, 256, 0, stream>>>(emb, enorm, counts);
    k_argmin <<< 128, 256, 0, stream>>>(x, emb, enorm, ws_idx, counts);
    k_zero   <<<4096, 256, 0, stream>>>((float2*)(out + OFF_ENC), 16777216 / 2);
    k_scatter<<<  64, 256, 0, stream>>>(ws_idx, out);
    k_quant  <<<8192, 256, 0, stream>>>(x, emb, ws_idx, out, partial);
    k_final  <<<   1, 256, 0, stream>>>(partial, counts, out);
}